// SuperGATConv_936302871048
// MI455X (gfx1250) — compile-verified
//
#include <hip/hip_runtime.h>
#include <math.h>

// ---------------- problem constants (match reference) ----------------
#define NN      50000
#define EE      1000000
#define IN_CH   128
#define HEADS   4
#define OUT_C   16
#define HID     64          // HEADS*OUT_C
#define NEG_SLOPE 0.2f

typedef __attribute__((ext_vector_type(2))) float v2f;
typedef __attribute__((ext_vector_type(8))) float v8f;

// order-preserving float->uint encode for atomicMax-based segment max
__device__ __forceinline__ unsigned enc_ord(float f) {
    unsigned b = __float_as_uint(f);
    return (b & 0x80000000u) ? ~b : (b | 0x80000000u);
}
__device__ __forceinline__ float dec_ord(unsigned u) {
    return __uint_as_float((u & 0x80000000u) ? (u ^ 0x80000000u) : ~u);
}
__device__ __forceinline__ float lrelu(float x) {
    return x > 0.0f ? x : x * NEG_SLOPE;
}

// ---------------- K0: init out=bias, m=0 (== -inf encoded), s=0 ------
__global__ void k_init(float* __restrict__ out, const float* __restrict__ bias,
                       unsigned* __restrict__ m, float* __restrict__ s) {
    int i = blockIdx.x * blockDim.x + threadIdx.x;
    if (i < NN * HID)   out[i] = bias[i & (HID - 1)];
    if (i < NN * HEADS) { m[i] = 0u; s[i] = 0.0f; }
}

// ---------------- K1: proj = x @ W^T via V_WMMA_F32_16X16X4_F32 ------
// block = 128 threads = 4 waves; wave w computes the 16x16 tile for
// head w (channels [16w,16w+16)) of a 16-row node strip. 32 WMMAs over K=128.
__global__ __launch_bounds__(128)
void k_gemm_wmma(const float* __restrict__ x, const float* __restrict__ W,
                 float* __restrict__ proj) {
    const int lane = threadIdx.x & 31;
    const int wave = threadIdx.x >> 5;      // 0..3 -> N tile / head
    const int m0   = blockIdx.x * 16;       // node row base
    const int n0   = wave * 16;             // output channel base
    const int half = lane >> 4;             // 0: K=0,1  1: K=2,3
    const int l    = lane & 15;

    // A: 16x4 f32 frag — lane l(+16*half) holds x[m0+l][kk + 2*half + {0,1}]
    const float* aP = x + (m0 + l) * IN_CH + half * 2;
    // B: 4x16 f32 frag — B[k][n] = W[n][k]; lane holds W[n0+l][kk + 2*half + {0,1}]
    const float* bP = W + (n0 + l) * IN_CH + half * 2;

    v8f c = {};
#pragma unroll 4
    for (int kk = 0; kk < IN_CH; kk += 4) {
        v2f a, b;
        a.x = aP[kk]; a.y = aP[kk + 1];
        b.x = bP[kk]; b.y = bP[kk + 1];
        c = __builtin_amdgcn_wmma_f32_16x16x4_f32(
                /*neg_a=*/false, a, /*neg_b=*/false, b,
                /*c_mod=*/(short)0, c, /*reuse_a=*/false, /*reuse_b=*/false);
    }
    // C/D layout: VGPR r => row M = r + 8*half, col N = l
    float* o = proj + (m0 + half * 8) * HID + n0 + l;
#pragma unroll
    for (int r = 0; r < 8; ++r) o[r * HID] = c[r];
}

// ---------------- K2: per-node attention logit halves ---------------
__global__ void k_alpha(const float* __restrict__ proj,
                        const float* __restrict__ att_src,
                        const float* __restrict__ att_dst,
                        float* __restrict__ asrc, float* __restrict__ adst) {
    int i = blockIdx.x * blockDim.x + threadIdx.x;      // n*HEADS + h
    if (i >= NN * HEADS) return;
    int h = i & (HEADS - 1);
    const float* p  = proj + (i >> 2) * HID + h * OUT_C;
    const float* as = att_src + h * OUT_C;
    const float* ad = att_dst + h * OUT_C;
    float s1 = 0.0f, s2 = 0.0f;
#pragma unroll
    for (int c = 0; c < OUT_C; ++c) { float v = p[c]; s1 += v * as[c]; s2 += v * ad[c]; }
    asrc[i] = s1; adst[i] = s2;
}

// helper: edge -> (row, col), self-loops appended after the E real edges
__device__ __forceinline__ void edge_rc(const int* ei, int e, int& r, int& c) {
    if (e < EE) { r = ei[e]; c = ei[EE + e]; }
    else        { r = c = e - EE; }
}

// ---------------- K3: segment max over destination ------------------
__global__ void k_seg_max(const int* __restrict__ ei,
                          const float* __restrict__ asrc,
                          const float* __restrict__ adst,
                          unsigned* __restrict__ m) {
    int i = blockIdx.x * blockDim.x + threadIdx.x;      // (E+N)*HEADS
    if (i >= (EE + NN) * HEADS) return;
    int e = i >> 2, h = i & (HEADS - 1);
    int r, c; edge_rc(ei, e, r, c);
    float lg = lrelu(asrc[r * HEADS + h] + adst[c * HEADS + h]);
    atomicMax(&m[c * HEADS + h], enc_ord(lg));
}

// ---------------- K4: segment sum of exp(logit - max) ---------------
__global__ void k_seg_sum(const int* __restrict__ ei,
                          const float* __restrict__ asrc,
                          const float* __restrict__ adst,
                          const unsigned* __restrict__ m,
                          float* __restrict__ s) {
    int i = blockIdx.x * blockDim.x + threadIdx.x;
    if (i >= (EE + NN) * HEADS) return;
    int e = i >> 2, h = i & (HEADS - 1);
    int r, c; edge_rc(ei, e, r, c);
    float lg = lrelu(asrc[r * HEADS + h] + adst[c * HEADS + h]);
    float ev = __expf(lg - dec_ord(m[c * HEADS + h]));
    atomicAdd(&s[c * HEADS + h], ev);
}

// ---------------- K5: weighted scatter-add aggregation --------------
// one thread per (edge, channel k in [0,64)); consecutive threads cover one
// edge's 64 channels -> coalesced proj reads / out atomics (L2-resident).
__global__ void k_aggregate(const int* __restrict__ ei,
                            const float* __restrict__ proj,
                            const float* __restrict__ asrc,
                            const float* __restrict__ adst,
                            const unsigned* __restrict__ m,
                            const float* __restrict__ s,
                            float* __restrict__ out) {
    long long i = (long long)blockIdx.x * blockDim.x + threadIdx.x;
    if (i >= (long long)(EE + NN) * HID) return;
    int e = (int)(i >> 6), k = (int)(i & (HID - 1)), h = k >> 4;
    int r, c; edge_rc(ei, e, r, c);
    float lg = lrelu(asrc[r * HEADS + h] + adst[c * HEADS + h]);
    float w  = __expf(lg - dec_ord(m[c * HEADS + h])) / s[c * HEADS + h];
    atomicAdd(&out[c * HID + k], proj[r * HID + k] * w);
}

// ---------------- launch ----------------
extern "C" void kernel_launch(void* const* d_in, const int* in_sizes, int n_in,
                              void* d_out, int out_size, void* d_ws, size_t ws_size,
                              hipStream_t stream) {
    const float* x       = (const float*)d_in[0];   // [N,128]
    const int*   ei      = (const int*)  d_in[1];   // [2,E]
    const float* W       = (const float*)d_in[2];   // [64,128]
    const float* att_src = (const float*)d_in[3];   // [4,16]
    const float* att_dst = (const float*)d_in[4];   // [4,16]
    const float* bias    = (const float*)d_in[5];   // [64]
    float* out = (float*)d_out;                     // [N,64]

    // workspace layout (floats): proj | asrc | adst | m(u32) | s  == N*80 = 16 MB
    float*    proj = (float*)d_ws;
    float*    asrc = proj + (size_t)NN * HID;
    float*    adst = asrc + (size_t)NN * HEADS;
    unsigned* m    = (unsigned*)(adst + (size_t)NN * HEADS);
    float*    s    = (float*)(m + (size_t)NN * HEADS);

    const int B = 256;
    // K0: init out/m/s
    k_init<<<(NN * HID + B - 1) / B, B, 0, stream>>>(out, bias, m, s);
    // K1: f32 WMMA GEMM, 3125 strips x 4 waves
    k_gemm_wmma<<<NN / 16, 128, 0, stream>>>(x, W, proj);
    // K2: per-node alpha halves
    k_alpha<<<(NN * HEADS + B - 1) / B, B, 0, stream>>>(proj, att_src, att_dst, asrc, adst);
    // K3/K4: segment softmax stats over destinations
    int et = (EE + NN) * HEADS;
    k_seg_max<<<(et + B - 1) / B, B, 0, stream>>>(ei, asrc, adst, m);
    k_seg_sum<<<(et + B - 1) / B, B, 0, stream>>>(ei, asrc, adst, m, s);
    // K5: weighted scatter-add
    long long tot = (long long)(EE + NN) * HID;
    k_aggregate<<<(unsigned)((tot + B - 1) / B), B, 0, stream>>>(ei, proj, asrc, adst, m, s, out);
}